// SOM_45389214384311
// MI455X (gfx1250) — compile-verified
//
#include <hip/hip_runtime.h>
#include <hip/hip_bf16.h>

// ---------------------------------------------------------------------------
// SOM neighbourhood function for MI455X (gfx1250, wave32, WMMA).
//   dists = ||x||^2 - 2 x.wT + ||w||^2 ; bmu = argmin_n dists
//   out[b][n] = exp(-grid_dist2(n, bmu[b]) / r^2)
// ||x||^2 is row-constant -> argmin of (w_sq[n] - 2*dot[m][n]); the 64MB
// distance matrix is never materialized. Dot products via
// V_WMMA_F32_16X16X4_F32. Global->LDS staging uses CDNA5 async-to-LDS
// (ASYNCcnt) with double-buffered LDS stages so DMA overlaps WMMA compute.
// ---------------------------------------------------------------------------

typedef __attribute__((ext_vector_type(2))) float v2f;
typedef __attribute__((ext_vector_type(8))) float v8f;
typedef __attribute__((ext_vector_type(4))) int   v4i;

#define SOM_B 4096
#define SOM_D 1024
#define SOM_N 4096

#define TM 64            // block tile M
#define TN 128           // block tile N
#define TK 32            // K per LDS stage
#define LDST 36          // LDS row stride (floats): padded, 16B-aligned rows
#define NKB (SOM_D / TK) // 32 K-blocks

// ---- CDNA5 async global->LDS copy (ASYNCcnt) ------------------------------
#if defined(__has_builtin)
#  if __has_builtin(__builtin_amdgcn_global_load_async_to_lds_b128) && \
      __has_builtin(__builtin_amdgcn_s_wait_asynccnt)
#    define SOM_ASYNC 1
#  endif
#endif
#ifndef SOM_ASYNC
#  define SOM_ASYNC 0
#endif

#if SOM_ASYNC
typedef __attribute__((address_space(1))) v4i som_gv4i;  // global int4
typedef __attribute__((address_space(3))) v4i som_lv4i;  // LDS int4
__device__ __forceinline__ void som_cp16(const float* g, float* l) {
    // 16B per lane, lands directly in LDS, tracked by ASYNCcnt.
    __builtin_amdgcn_global_load_async_to_lds_b128(
        (som_gv4i*)(unsigned long long)g,
        (som_lv4i*)(unsigned long long)l, 0, 0);
}
#define SOM_WAIT_ASYNC(n) __builtin_amdgcn_s_wait_asynccnt(n)
#else
__device__ __forceinline__ void som_cp16(const float* g, float* l) {
    *(float4*)l = *(const float4*)g;
}
#define SOM_WAIT_ASYNC(n) ((void)0)
#endif

// ---------------- kernel 1: w_sq[n] = sum_k w[n][k]^2 ----------------------
__global__ __launch_bounds__(256)
void som_wsq_kernel(const float* __restrict__ w, float* __restrict__ wsq) {
    const int n = blockIdx.x;
    const float* row = w + (size_t)n * SOM_D;
    float s = 0.0f;
    for (int k = threadIdx.x; k < SOM_D; k += 256) {
        float v = row[k];
        s += v * v;
    }
    __shared__ float red[256];
    red[threadIdx.x] = s;
    __syncthreads();
    for (int off = 128; off > 0; off >>= 1) {
        if ((int)threadIdx.x < off) red[threadIdx.x] += red[threadIdx.x + off];
        __syncthreads();
    }
    if (threadIdx.x == 0) wsq[n] = red[0];
}

// ---------------- kernel 2: init packed BMU table --------------------------
__global__ __launch_bounds__(256)
void som_init_kernel(unsigned long long* __restrict__ bmu) {
    int i = blockIdx.x * 256 + threadIdx.x;
    if (i < SOM_B) bmu[i] = ~0ull;
}

// Monotonic unsigned key for float ordering (ascending).
__device__ __forceinline__ unsigned som_fkey(float f) {
    unsigned u = __float_as_uint(f);
    return (u & 0x80000000u) ? ~u : (u | 0x80000000u);
}

// ---------------- kernel 3: WMMA GEMM + fused per-row argmin ---------------
// Stage issue: 2 A-copies + 4 B-copies per thread = 6 async instrs per wave.
__device__ __forceinline__ void som_issue_stage(const float* __restrict__ X,
                                                const float* __restrict__ W,
                                                int bm, int bn, int k0,
                                                float* __restrict__ As,
                                                float* __restrict__ Bs,
                                                int tid) {
    #pragma unroll
    for (int i = 0; i < 2; ++i) {
        int idx = tid + i * 256;
        int r = idx >> 3, c4 = (idx & 7) << 2;
        som_cp16(X + (size_t)(bm + r) * SOM_D + k0 + c4, As + r * LDST + c4);
    }
    #pragma unroll
    for (int i = 0; i < 4; ++i) {
        int idx = tid + i * 256;
        int r = idx >> 3, c4 = (idx & 7) << 2;
        som_cp16(W + (size_t)(bn + r) * SOM_D + k0 + c4, Bs + r * LDST + c4);
    }
}

__global__ __launch_bounds__(256)
void som_gemm_argmin_kernel(const float* __restrict__ X,   // [B, D]
                            const float* __restrict__ W,   // [N, D]
                            const float* __restrict__ wsq, // [N]
                            unsigned long long* __restrict__ bmu) { // [B]
    __shared__ float As[2][TM * LDST];  // double-buffered [m][k]
    __shared__ float Bs[2][TN * LDST];  // double-buffered [n][k]

    const int bm = blockIdx.y * TM;
    const int bn = blockIdx.x * TN;
    const int tid  = threadIdx.x;
    const int wave = tid >> 5;        // 0..7
    const int lane = tid & 31;
    const int wm = wave >> 2;         // 0..1 : wave row tile (32 rows)
    const int wn = wave & 3;          // 0..3 : wave col tile (32 cols)

    const int lhalf = lane & 15;          // M (A) / N (B,D) index within tile
    const int khalf = (lane >> 4) << 1;   // K pair select: 0 or 2

    v8f acc[2][2] = {};                    // 2x2 grid of 16x16 f32 tiles

    // Prologue: stage 0 in flight.
    som_issue_stage(X, W, bm, bn, 0, As[0], Bs[0], tid);

    for (int kb = 0; kb < NKB; ++kb) {
        const int cur = kb & 1;
        if (kb + 1 < NKB) {
            // Issue next stage, then wait for the current one (in-order:
            // waiting down to 6 outstanding completes the previous 6).
            som_issue_stage(X, W, bm, bn, (kb + 1) * TK,
                            As[cur ^ 1], Bs[cur ^ 1], tid);
            SOM_WAIT_ASYNC(6);
        } else {
            SOM_WAIT_ASYNC(0);
        }
        __syncthreads();   // all waves' stage-`cur` data landed in LDS

        const float* Ac = As[cur];
        const float* Bc = Bs[cur];
        #pragma unroll
        for (int kk = 0; kk < TK; kk += 4) {
            v2f af[2], bf[2];
            #pragma unroll
            for (int i = 0; i < 2; ++i) {
                int m = wm * 32 + i * 16 + lhalf;
                af[i] = *(const v2f*)(Ac + m * LDST + kk + khalf);
            }
            #pragma unroll
            for (int j = 0; j < 2; ++j) {
                int n = wn * 32 + j * 16 + lhalf;
                bf[j] = *(const v2f*)(Bc + n * LDST + kk + khalf);
            }
            #pragma unroll
            for (int i = 0; i < 2; ++i)
                #pragma unroll
                for (int j = 0; j < 2; ++j)
                    acc[i][j] = __builtin_amdgcn_wmma_f32_16x16x4_f32(
                        /*neg_a=*/false, af[i], /*neg_b=*/false, bf[j],
                        /*c_mod=*/(short)0, acc[i][j],
                        /*reuse_a=*/false, /*reuse_b=*/false);
        }
        __syncthreads();   // stage `cur` free for the copy issued next iter
    }

    // ---- fused argmin of (wsq[n] - 2*dot) over this wave's 32 columns ----
    // D layout: reg v of acc holds row (v + 8*(lane>=16)), col = lane&15.
    const int half = lane >> 4;
    unsigned nj0 = (unsigned)(bn + wn * 32 + 0 * 16 + lhalf);
    unsigned nj1 = (unsigned)(bn + wn * 32 + 1 * 16 + lhalf);
    const float wsq0 = wsq[nj0];
    const float wsq1 = wsq[nj1];

    #pragma unroll
    for (int i = 0; i < 2; ++i) {
        #pragma unroll
        for (int v = 0; v < 8; ++v) {
            float s0 = wsq0 - 2.0f * acc[i][0][v];
            float s1 = wsq1 - 2.0f * acc[i][1][v];
            unsigned long long p0 = ((unsigned long long)som_fkey(s0) << 32) | nj0;
            unsigned long long p1 = ((unsigned long long)som_fkey(s1) << 32) | nj1;
            unsigned long long p = p1 < p0 ? p1 : p0;
            // min-reduce across the 16-lane half (masks < 16 keep halves apart)
            #pragma unroll
            for (int m = 1; m < 16; m <<= 1) {
                unsigned long long q = __shfl_xor(p, m, 32);
                p = q < p ? q : p;
            }
            if (lhalf == 0) {
                int row = bm + wm * 32 + i * 16 + v + 8 * half;
                atomicMin(&bmu[row], p);
            }
        }
    }
}

// ---------------- kernel 4: neighbourhood = exp(-d2/r^2) -------------------
__global__ __launch_bounds__(256)
void som_neigh_kernel(const unsigned long long* __restrict__ bmu, // [B] packed
                      const int* __restrict__ loc,                // [N,2]
                      const int* __restrict__ rad,                // [1]
                      float* __restrict__ out) {                  // [B,N]
    const int n = blockIdx.x * 256 + threadIdx.x;
    const int b = blockIdx.y;
    const float r = (float)rad[0];
    const float inv_r2 = 1.0f / (r * r);
    const unsigned bi = (unsigned)(bmu[b] & 0xFFFFFFFFu);
    const int2 bl = ((const int2*)loc)[bi];
    const int2 nl = ((const int2*)loc)[n];
    const float dx = (float)(nl.x - bl.x);
    const float dy = (float)(nl.y - bl.y);
    out[(size_t)b * SOM_N + n] = __expf(-(dx * dx + dy * dy) * inv_r2);
}

// ---------------------------------------------------------------------------
extern "C" void kernel_launch(void* const* d_in, const int* in_sizes, int n_in,
                              void* d_out, int out_size, void* d_ws, size_t ws_size,
                              hipStream_t stream) {
    const float* batch   = (const float*)d_in[0];   // [4096,1024]
    const float* weights = (const float*)d_in[1];   // [4096,1024]
    const int*   loc     = (const int*)d_in[2];     // [4096,2]
    const int*   radius  = (const int*)d_in[3];     // [1]
    float* out = (float*)d_out;                     // [4096,4096]

    // workspace layout: [bmu packed u64 x 4096][wsq f32 x 4096]
    unsigned long long* bmu = (unsigned long long*)d_ws;
    float* wsq = (float*)((char*)d_ws + SOM_B * sizeof(unsigned long long));

    som_wsq_kernel<<<SOM_N, 256, 0, stream>>>(weights, wsq);
    som_init_kernel<<<(SOM_B + 255) / 256, 256, 0, stream>>>(bmu);

    dim3 ggrid(SOM_N / TN, SOM_B / TM);   // (32, 64)
    som_gemm_argmin_kernel<<<ggrid, 256, 0, stream>>>(batch, weights, wsq, bmu);

    dim3 ngrid(SOM_N / 256, SOM_B);       // (16, 4096)
    som_neigh_kernel<<<ngrid, 256, 0, stream>>>(bmu, loc, radius, out);
}